// GSAAttention_19791209300599
// MI455X (gfx1250) — compile-verified
//
#include <hip/hip_runtime.h>
#include <cstdint>
#include <cstddef>

// ---------------------------------------------------------------------------
// GSA attention for MI455X (gfx1250), fp32 end-to-end via V_WMMA_F32_16X16X4_F32.
//
// Roofline: ~50 GFLOP vs ~150 MB HBM traffic -> memory bound at 23.3 TB/s.
// fp32 WMMA keeps reference precision while using the matrix pipe.
//
// This revision removes the v_mov operand-packing seen in round-2 asm by
// making every WMMA B-operand K-pair contiguous in memory:
//   * staged W^T tile is pair-interleaved in LDS  -> single ds_load_b64
//   * Q^T workspace is pair-interleaved over n    -> single global_load_b64
// ---------------------------------------------------------------------------

#if defined(__HIP_DEVICE_COMPILE__)
#if !__has_builtin(__builtin_amdgcn_wmma_f32_16x16x4_f32)
#error "missing __builtin_amdgcn_wmma_f32_16x16x4_f32 on gfx1250 device pass"
#endif
#endif

typedef __attribute__((ext_vector_type(2))) float v2f;
typedef __attribute__((ext_vector_type(8))) float v8f;

static constexpr int kB = 8;     // batch
static constexpr int kN = 2048;  // tokens
static constexpr int kC = 512;   // channels
static constexpr int kCR = 64;   // reduced channels

__device__ __forceinline__ v8f wmma4(v2f a, v2f b, v8f c) {
#if defined(__HIP_DEVICE_COMPILE__)
  // D(16x16,f32) = A(16x4,f32) x B(4x16,f32) + C
  return __builtin_amdgcn_wmma_f32_16x16x4_f32(
      /*neg_a=*/false, a, /*neg_b=*/false, b,
      /*c_mod=*/(short)0, c, /*reuse_a=*/false, /*reuse_b=*/false);
#else
  (void)a; (void)b;
  return c;  // host pass never executes this; keep it parseable
#endif
}

__device__ __forceinline__ v8f v8zero() {
  v8f z = {0.f, 0.f, 0.f, 0.f, 0.f, 0.f, 0.f, 0.f};
  return z;
}

// ---------------------------------------------------------------------------
// Kernel 1: fused projections.  out[n, o] = sum_c x[n,c] * W[o,c] + b[o]
// Column groups (blockIdx.y): 0 -> K (Wk, 64), 1 -> Vq (Wv, 64, this IS v^T),
// 2..9 -> Q^T (Wq rows (g-2)*64..).  K/V stored row-major [n, o]; Q stored
// pair-interleaved over n: Qp[((n>>1)*C + c)*2 + (n&1)], which is exactly the
// B-operand layout the attention PV loop wants (one b64 per operand).
// Block: 256 thr (8 waves), tile 128 rows x 64 cols; wave -> 16 rows x 64 cols.
// ---------------------------------------------------------------------------
__global__ __launch_bounds__(256) void gsa_proj_kernel(
    const float* __restrict__ x,
    const float* __restrict__ Wk, const float* __restrict__ bk,
    const float* __restrict__ Wv, const float* __restrict__ bv,
    const float* __restrict__ Wq, const float* __restrict__ bq,
    float* __restrict__ Ko, float* __restrict__ Vo, float* __restrict__ Qp) {
  // sx: stride 36 -> b64 A-reads 8B-aligned, 16 consecutive rows -> 16 banks.
  // swp: pair-interleaved W^T: swp[(k>>1)*130 + o*2 + (k&1)]; B-read is one
  //      aligned b64, 16 lanes cover 32 consecutive dwords (conflict-free).
  __shared__ float sx[128 * 36];
  __shared__ float swp[16 * 130];

  const int tid = threadIdx.x;
  const int lane = tid & 31, wave = tid >> 5;
  const int ln = lane & 15, hi = lane >> 4, h0 = hi << 1;
  const int gr0 = blockIdx.x * 128;  // flat row in [0, B*N)
  const int g = blockIdx.y;

  const float* W;
  const float* bias;
  if (g == 0) {
    W = Wk; bias = bk;
  } else if (g == 1) {
    W = Wv; bias = bv;
  } else {
    W = Wq + (size_t)(g - 2) * 64 * kC;
    bias = bq + (g - 2) * 64;
  }

  v8f acc[4];
#pragma unroll
  for (int t = 0; t < 4; ++t) acc[t] = v8zero();

  for (int kb = 0; kb < kC; kb += 32) {
    // Stage X chunk [128 x 32], coalesced float4 loads.
#pragma unroll
    for (int i = 0; i < 4; ++i) {
      const int idx = tid + i * 256;        // 0..1023
      const int row = idx >> 3, q = idx & 7;
      float4 v = *(const float4*)&x[(size_t)(gr0 + row) * kC + kb + q * 4];
      *(float4*)&sx[row * 36 + q * 4] = v;
    }
    // Stage W^T chunk pair-interleaved: float4 of (k..k+3) -> two b64 stores.
#pragma unroll
    for (int i = 0; i < 2; ++i) {
      const int idx = tid + i * 256;        // 0..511
      const int o = idx >> 3, q = idx & 7;
      float4 v = *(const float4*)&W[(size_t)o * kC + kb + q * 4];
      v2f lo = {v.x, v.y}, hiv = {v.z, v.w};
      *(v2f*)&swp[(q * 2 + 0) * 130 + o * 2] = lo;
      *(v2f*)&swp[(q * 2 + 1) * 130 + o * 2] = hiv;
    }
    __syncthreads();

#pragma unroll
    for (int k2 = 0; k2 < 8; ++k2) {
      // A operand: lane holds row M=ln, K pair (low half K0..1, high half K2..3)
      v2f a = *(const v2f*)&sx[(wave * 16 + ln) * 36 + k2 * 4 + h0];
#pragma unroll
      for (int tc = 0; tc < 4; ++tc) {
        // B pair (k, k+1) with k = k2*4 + h0 (even) -> row (k2*2 + hi).
        v2f b = *(const v2f*)&swp[(k2 * 2 + hi) * 130 + (tc * 16 + ln) * 2];
        acc[tc] = wmma4(a, b, acc[tc]);
      }
    }
    __syncthreads();
  }

  // Bias + store (D layout: element r -> row M = r + 8*hi, col N = ln).
#pragma unroll
  for (int tc = 0; tc < 4; ++tc) {
    const float bb = bias[tc * 16 + ln];
#pragma unroll
    for (int r = 0; r < 8; ++r) {
      const int row = gr0 + wave * 16 + r + 8 * hi;
      const float val = acc[tc][r] + bb;
      if (g == 0) {
        Ko[(size_t)row * kCR + tc * 16 + ln] = val;
      } else if (g == 1) {
        Vo[(size_t)row * kCR + tc * 16 + ln] = val;
      } else {
        const int c = (g - 2) * 64 + tc * 16 + ln;
        Qp[((size_t)(row >> 1) * kC + c) * 2 + (row & 1)] = val;
      }
    }
  }
}

// ---------------------------------------------------------------------------
// Kernel 2: transposed attention, flash-style online softmax over n.
//   queries  = Vo rows  [N x 64]
//   keys     = Ko rows  [N x 64]
//   values   = Qp       [N x 512], pair-interleaved over n
//   out[m,c] = gamma * (sum_n softmax_n(Ko[n]·Vo[m]) * Q^T[n,c]) / l_m + x[m,c]
// Block: 256 thr (8 waves) = (2 m-halves of 16) x (4 c-slices of 128).
// Grid: (N/32, B).
// ---------------------------------------------------------------------------
__global__ __launch_bounds__(256) void gsa_attn_kernel(
    const float* __restrict__ Ko, const float* __restrict__ Vo,
    const float* __restrict__ Qp, const float* __restrict__ x,
    const float* __restrict__ gammaP, float* __restrict__ out) {
  __shared__ float sK[64 * 68];  // K chunk [64 n][64 r], stride 68 (16B rows)
  __shared__ float sP[32 * 66];  // scores/probs [32 m][64 n], stride 66
  __shared__ float smax[32], ssum[32], sscale[32], sinv[32];

  const int tid = threadIdx.x;
  const int lane = tid & 31, wave = tid >> 5;
  const int ln = lane & 15, hi = lane >> 4, h0 = hi << 1;
  const int mh = wave >> 2;   // which 16-row m-half
  const int cq = wave & 3;    // which 128-wide c-slice (also 16-wide n-slice)
  const int b = blockIdx.y;
  const int m0 = blockIdx.x * 32;
  const size_t bN = (size_t)b * kN;

  if (tid < 32) {
    smax[tid] = -3.0e38f;
    ssum[tid] = 0.0f;
  }

  // Query tile A-operands resident in registers: Vo[m0 + mh*16 + ln][0..63].
  v2f vq[16];
#pragma unroll
  for (int s = 0; s < 16; ++s)
    vq[s] = *(const v2f*)&Vo[(bN + m0 + mh * 16 + ln) * kCR + s * 4 + h0];

  v8f acc[8];  // 16 m x 128 c output accumulator
#pragma unroll
  for (int t = 0; t < 8; ++t) acc[t] = v8zero();

  for (int n0 = 0; n0 < kN; n0 += 64) {
    // ---- stage K chunk (coalesced) -------------------------------------
#pragma unroll
    for (int i = 0; i < 4; ++i) {
      const int idx = tid + i * 256;        // 0..1023
      const int row = idx >> 4, q = idx & 15;
      float4 v = *(const float4*)&Ko[(bN + n0 + row) * kCR + q * 4];
      *(float4*)&sK[row * 68 + q * 4] = v;
    }
    if (n0 + 64 < kN)  // hint next K chunk into cache (global_prefetch_b8)
      __builtin_prefetch(&Ko[(bN + n0 + 64) * kCR] + tid * 16, 0, 1);
    __syncthreads();

    // ---- S piece: wave computes S[mh-half 16m x 16n (cq)] --------------
    v8f s = v8zero();
#pragma unroll
    for (int ks = 0; ks < 16; ++ks) {
      // B = K^T: B[r][n] = Ko[n][r]; lane col n = cq*16+ln, K pair via h0.
      v2f bk2 = *(const v2f*)&sK[(cq * 16 + ln) * 68 + ks * 4 + h0];
      s = wmma4(vq[ks], bk2, s);
    }
#pragma unroll
    for (int r = 0; r < 8; ++r)
      sP[(mh * 16 + r + 8 * hi) * 66 + cq * 16 + ln] = s[r];
    __syncthreads();

    // ---- online softmax stats per m-row --------------------------------
    if (tid < 32) {
      float mx = -3.0e38f;
      for (int j = 0; j < 64; ++j) mx = fmaxf(mx, sP[tid * 66 + j]);
      const float nm = fmaxf(smax[tid], mx);
      const float sc = __expf(smax[tid] - nm);
      float rs = 0.0f;
      for (int j = 0; j < 64; ++j) {
        const float p = __expf(sP[tid * 66 + j] - nm);
        sP[tid * 66 + j] = p;
        rs += p;
      }
      ssum[tid] = ssum[tid] * sc + rs;
      smax[tid] = nm;
      sscale[tid] = sc;
    }
    __syncthreads();

    // ---- rescale accumulators ------------------------------------------
    float f[8];
#pragma unroll
    for (int r = 0; r < 8; ++r) f[r] = sscale[mh * 16 + r + 8 * hi];
#pragma unroll
    for (int tc = 0; tc < 8; ++tc)
#pragma unroll
      for (int r = 0; r < 8; ++r) acc[tc][r] *= f[r];

    // ---- acc += P[16m x 64n] * Q^T[64n x 128c] -------------------------
    // B operand (Q^T[n][c], Q^T[n+1][c]) is contiguous in Qp: one b64 load,
    // 16 lanes -> 128 contiguous bytes.
#pragma unroll 4
    for (int ks = 0; ks < 16; ++ks) {
      v2f a = *(const v2f*)&sP[(mh * 16 + ln) * 66 + ks * 4 + h0];
      const int n = n0 + ks * 4 + h0;  // even
      const float* qrow = &Qp[((size_t)(n >> 1) * kC + cq * 128 + ln) * 2];
#pragma unroll
      for (int tc = 0; tc < 8; ++tc) {
        v2f bq2 = *(const v2f*)&qrow[tc * 32];  // (tc*16 cols) * 2
        acc[tc] = wmma4(a, bq2, acc[tc]);
      }
    }
    // next iteration's staging barrier also fences sP reuse
  }

  if (tid < 32) sinv[tid] = 1.0f / ssum[tid];
  __syncthreads();

  const float gm = gammaP[0];
#pragma unroll
  for (int tc = 0; tc < 8; ++tc) {
    const int c = cq * 128 + tc * 16 + ln;
#pragma unroll
    for (int r = 0; r < 8; ++r) {
      const int ml = mh * 16 + r + 8 * hi;
      const size_t m = bN + m0 + ml;
      out[m * kC + c] = gm * acc[tc][r] * sinv[ml] + x[m * kC + c];
    }
  }
}

// ---------------------------------------------------------------------------
// Host launcher. Workspace layout (floats):
//   Ko [B*N*64] | Vo [B*N*64] | Qp [B*N*512]  = 40 MB total.
// ---------------------------------------------------------------------------
extern "C" void kernel_launch(void* const* d_in, const int* in_sizes, int n_in,
                              void* d_out, int out_size, void* d_ws,
                              size_t ws_size, hipStream_t stream) {
  (void)in_sizes; (void)n_in; (void)out_size; (void)ws_size;
  const float* x = (const float*)d_in[0];
  const float* Wk = (const float*)d_in[1];
  const float* bk = (const float*)d_in[2];
  const float* Wv = (const float*)d_in[3];
  const float* bv = (const float*)d_in[4];
  const float* Wq = (const float*)d_in[5];
  const float* bq = (const float*)d_in[6];
  const float* gamma = (const float*)d_in[7];

  float* ws = (float*)d_ws;
  float* Ko = ws;                                   // [B*N, 64]
  float* Vo = ws + (size_t)kB * kN * kCR;           // [B*N, 64]
  float* Qp = ws + (size_t)2 * kB * kN * kCR;       // [B*N, 512] pair-interleaved

  dim3 gProj(kB * kN / 128, 10);  // 10 = (64+64+512)/64 output column groups
  gsa_proj_kernel<<<gProj, 256, 0, stream>>>(x, Wk, bk, Wv, bv, Wq, bq, Ko, Vo,
                                             Qp);

  dim3 gAttn(kN / 32, kB);
  gsa_attn_kernel<<<gAttn, 256, 0, stream>>>(Ko, Vo, Qp, x, gamma,
                                             (float*)d_out);
}